// MCGRUEncoder_5188320493812
// MI455X (gfx1250) — compile-verified
//
#include <hip/hip_runtime.h>
#include <hip/hip_bf16.h>

#define B_  32
#define T_  2048
#define D_  64
#define F_  8
#define H_  128
#define S_  32
#define K2_ 512      // D*F
#define KO_ 640      // H + D*F

typedef __attribute__((ext_vector_type(2))) float v2f;
typedef __attribute__((ext_vector_type(8))) float v8f;
typedef int v4i __attribute__((vector_size(16)));   // matches builtin param type

__device__ __forceinline__ float fast_sigmoid(float x) {
    return 1.0f / (1.0f + __expf(-x));
}
__device__ __forceinline__ float fast_tanh(float x) {
    return 1.0f - 2.0f / (__expf(2.0f * x) + 1.0f);
}

// --- CDNA5 async copy: global -> LDS, 16B per lane, tracked by ASYNCcnt ----
__device__ __forceinline__ void async_copy16(const void* gsrc, void* ldst) {
#if __has_builtin(__builtin_amdgcn_global_load_async_to_lds_b128)
    __builtin_amdgcn_global_load_async_to_lds_b128(
        (__attribute__((address_space(1))) v4i*)(uintptr_t)gsrc,
        (__attribute__((address_space(3))) v4i*)(uintptr_t)ldst, 0, 0);
#else
    unsigned lds = (unsigned)(uintptr_t)ldst;           // low 32b = LDS offset
    unsigned long long ga = (unsigned long long)(uintptr_t)gsrc;
    asm volatile("global_load_async_to_lds_b128 %0, %1, off"
                 :: "v"(lds), "v"(ga) : "memory");
#endif
}
__device__ __forceinline__ void wait_async0() {
#if __has_builtin(__builtin_amdgcn_s_wait_asynccnt)
    __builtin_amdgcn_s_wait_asynccnt(0);
#else
    asm volatile("s_wait_asynccnt 0x0" ::: "memory");
#endif
}

// ---------------------------------------------------------------------------
// Kernel 1: in_proj  y = x @ W + b, scattered to transposed layout
//   xp_t[(e*T + t)*B + b] so each GRU wave reads one coalesced 128B line/step.
//   Grid: 4096 m-tiles of 16 rows; block = 128 threads (4 waves cover N=64).
// ---------------------------------------------------------------------------
__global__ __launch_bounds__(128) void k_inproj(const float* __restrict__ x,
                                                const float* __restrict__ W,
                                                const float* __restrict__ bias,
                                                float* __restrict__ xp_t) {
    __shared__ __align__(16) float As[16 * 64];
    const int tid = threadIdx.x;
    const int r0  = blockIdx.x * 16;

    // Async-stage A tile (16 x 64 fp32 = 4KB) straight into LDS.
    const float4* xv = (const float4*)(x + (size_t)r0 * D_);
    #pragma unroll
    for (int i = 0; i < 2; ++i) {
        int idx = tid + i * 128;
        async_copy16(xv + idx, ((float4*)As) + idx);
    }
    wait_async0();
    __syncthreads();

    const int wave = tid >> 5, lane = tid & 31;
    const int n0 = wave * 16, lm = lane & 15, lh = lane >> 4;

    v8f acc = {};
    #pragma unroll 4
    for (int k = 0; k < 64; k += 4) {
        v2f a = *(const v2f*)&As[lm * 64 + k + 2 * lh];
        v2f b;
        b.x = W[(k + 2 * lh + 0) * D_ + n0 + lm];
        b.y = W[(k + 2 * lh + 1) * D_ + n0 + lm];
        acc = __builtin_amdgcn_wmma_f32_16x16x4_f32(false, a, false, b,
                                                    (short)0, acc, false, false);
    }

    const float bi = bias[n0 + lm];
    const int  b_  = r0 / T_;         // tile never crosses a batch boundary
    const int  t0  = r0 % T_;
    const int  e   = n0 + lm;
    #pragma unroll
    for (int v = 0; v < 8; ++v) {
        int t = t0 + v + 8 * lh;
        xp_t[((size_t)e * T_ + t) * B_ + b_] = acc[v] + bi;
    }
}

// ---------------------------------------------------------------------------
// Kernel 2: fused GRU scan. One wave32 per channel d; lane = batch b.
//   whh (transposed [f][g]) in LDS -> broadcast float4 reads, no conflicts.
//   PyTorch gate order r,z,n;  h' = n + z*(h - n).
// ---------------------------------------------------------------------------
__global__ __launch_bounds__(32) void k_gru(const float* __restrict__ xp_t,
                                            const float* __restrict__ wih_g,
                                            const float* __restrict__ whh_g,
                                            const float* __restrict__ bih_g,
                                            const float* __restrict__ bhh_g,
                                            float* __restrict__ gru_out) {
    const int d    = blockIdx.x;
    const int lane = threadIdx.x;          // = batch b

    __shared__ __align__(16) float Ws[8][24];   // Ws[f][g] = whh[d][g][f]
    for (int i = lane; i < 192; i += 32) {
        int f = i / 24, g = i % 24;
        Ws[f][g] = whh_g[((size_t)d * 24 + g) * 8 + f];
    }
    __syncthreads();

    float wih[24], bih[24], bhh[24];
    #pragma unroll
    for (int g = 0; g < 24; ++g) {
        wih[g] = wih_g[d * 24 + g];
        bih[g] = bih_g[d * 24 + g];
        bhh[g] = bhh_g[d * 24 + g];
    }

    float h[8];
    #pragma unroll
    for (int j = 0; j < 8; ++j) h[j] = 0.0f;

    const float* xcol = xp_t + (size_t)d * T_ * B_;
    float*       outp = gru_out + (size_t)lane * T_ * K2_ + d * 8;

    for (int t = 0; t < T_; ++t) {
        const float xt = xcol[t * B_ + lane];

        float acc[24];
        #pragma unroll
        for (int g = 0; g < 24; ++g) acc[g] = bhh[g];
        #pragma unroll
        for (int f = 0; f < 8; ++f) {
            const float hf = h[f];
            #pragma unroll
            for (int q = 0; q < 6; ++q) {
                float4 w = *(const float4*)&Ws[f][4 * q];
                acc[4 * q + 0] += hf * w.x;
                acc[4 * q + 1] += hf * w.y;
                acc[4 * q + 2] += hf * w.z;
                acc[4 * q + 3] += hf * w.w;
            }
        }

        #pragma unroll
        for (int j = 0; j < 8; ++j) {
            float rg = fast_sigmoid(xt * wih[j]      + bih[j]      + acc[j]);
            float zg = fast_sigmoid(xt * wih[8 + j]  + bih[8 + j]  + acc[8 + j]);
            float ng = fast_tanh   (xt * wih[16 + j] + bih[16 + j] + rg * acc[16 + j]);
            h[j] = ng + zg * (h[j] - ng);
        }

        float4 lo = {h[0], h[1], h[2], h[3]};
        float4 hi = {h[4], h[5], h[6], h[7]};
        *(float4*)(outp + (size_t)t * K2_)     = lo;
        *(float4*)(outp + (size_t)t * K2_ + 4) = hi;
    }
}

// ---------------------------------------------------------------------------
// Kernel 3: static path.
//   s_contrib[b][h] = out_b[h] + (static@static_w + static_b) @ out_w[0:H]
// ---------------------------------------------------------------------------
__global__ __launch_bounds__(256) void k_static(const float* __restrict__ stat,
                                                const float* __restrict__ sw,
                                                const float* __restrict__ sb,
                                                const float* __restrict__ out_w,
                                                const float* __restrict__ out_b,
                                                float* __restrict__ s_contrib) {
    __shared__ float Sf[B_ * H_];
    const int tid = threadIdx.x;
    for (int i = tid; i < B_ * H_; i += 256) {
        int b = i / H_, hh = i % H_;
        float a = sb[hh];
        for (int j = 0; j < S_; ++j) a += stat[b * S_ + j] * sw[j * H_ + hh];
        Sf[i] = a;
    }
    __syncthreads();
    for (int i = tid; i < B_ * H_; i += 256) {
        int b = i / H_, hh = i % H_;
        float a = out_b[hh];
        for (int j = 0; j < H_; ++j) a += Sf[b * H_ + j] * out_w[j * H_ + hh];
        s_contrib[i] = a;
    }
}

// One-time transpose: Wt[n][k] = out_w[H + k][n]  (128 x 512, 256KB, L2-hot)
__global__ __launch_bounds__(256) void k_wt(const float* __restrict__ out_w,
                                            float* __restrict__ Wt) {
    int idx = blockIdx.x * 256 + threadIdx.x;          // 256 blocks
    int n = idx % H_, k = idx / H_;
    Wt[(size_t)n * K2_ + k] = out_w[(size_t)(H_ + k) * H_ + n];
}

// ---------------------------------------------------------------------------
// Kernel 4: main GEMM  out[r][h] = s_contrib[r/T][h] + gru[r][:] @ Wt[h][:]
//   Grid: 2048 blocks of 32 rows; block = 256 threads = 8 waves (N=128).
//   A (32 x 512 = 64KB) async-staged into LDS; 2 m-tiles per wave amortize B.
//   Inner 4-k step: 1 global b64 (Wt) + 2 LDS b64 + 2 wmma.
// ---------------------------------------------------------------------------
__global__ __launch_bounds__(256) void k_out(const float* __restrict__ gru,
                                             const float* __restrict__ Wt,
                                             const float* __restrict__ s_contrib,
                                             float* __restrict__ out) {
    __shared__ __align__(16) float As[32 * K2_];   // 64KB
    const int tid = threadIdx.x;
    const int r0  = blockIdx.x * 32;

    const float4* gv = (const float4*)(gru + (size_t)r0 * K2_);
    #pragma unroll
    for (int i = 0; i < 16; ++i) {                 // 4096 float4 / 256 threads
        int idx = tid + i * 256;
        async_copy16(gv + idx, ((float4*)As) + idx);
    }
    wait_async0();
    __syncthreads();

    const int wave = tid >> 5, lane = tid & 31;
    const int n0 = wave * 16, lm = lane & 15, lh = lane >> 4;
    const float* wrow = Wt + (size_t)(n0 + lm) * K2_;

    v8f acc0 = {}, acc1 = {};
    #pragma unroll 4
    for (int k = 0; k < K2_; k += 4) {
        v2f b  = *(const v2f*)&wrow[k + 2 * lh];
        v2f a0 = *(const v2f*)&As[(     lm) * K2_ + k + 2 * lh];
        v2f a1 = *(const v2f*)&As[(16 + lm) * K2_ + k + 2 * lh];
        acc0 = __builtin_amdgcn_wmma_f32_16x16x4_f32(false, a0, false, b,
                                                     (short)0, acc0, false, false);
        acc1 = __builtin_amdgcn_wmma_f32_16x16x4_f32(false, a1, false, b,
                                                     (short)0, acc1, false, false);
    }

    const int   b_ = r0 / T_;                      // 32 | T, same batch per block
    const float sc = s_contrib[b_ * H_ + n0 + lm];
    #pragma unroll
    for (int v = 0; v < 8; ++v) {
        out[(size_t)(r0 +      v + 8 * lh) * H_ + n0 + lm] = acc0[v] + sc;
        out[(size_t)(r0 + 16 + v + 8 * lh) * H_ + n0 + lm] = acc1[v] + sc;
    }
}

// ---------------------------------------------------------------------------
extern "C" void kernel_launch(void* const* d_in, const int* in_sizes, int n_in,
                              void* d_out, int out_size, void* d_ws, size_t ws_size,
                              hipStream_t stream) {
    (void)in_sizes; (void)n_in; (void)out_size; (void)ws_size;

    const float* x        = (const float*)d_in[0];
    const float* stat     = (const float*)d_in[1];
    const float* in_w     = (const float*)d_in[2];
    const float* in_b     = (const float*)d_in[3];
    const float* gru_wih  = (const float*)d_in[4];
    const float* gru_whh  = (const float*)d_in[5];
    const float* gru_bih  = (const float*)d_in[6];
    const float* gru_bhh  = (const float*)d_in[7];
    const float* static_w = (const float*)d_in[8];
    const float* static_b = (const float*)d_in[9];
    const float* out_w    = (const float*)d_in[10];
    const float* out_b    = (const float*)d_in[11];
    float*       out      = (float*)d_out;

    // workspace partition
    char* ws = (char*)d_ws;
    size_t off = 0;
    float* xp_t      = (float*)(ws + off); off += (size_t)D_ * T_ * B_ * 4;   // 16 MiB
    float* gru_buf   = (float*)(ws + off); off += (size_t)B_ * T_ * K2_ * 4;  // 128 MiB
    float* s_contrib = (float*)(ws + off); off += (size_t)B_ * H_ * 4;        // 16 KiB
    float* Wt        = (float*)(ws + off);                                    // 256 KiB

    k_wt    <<<(H_ * K2_) / 256, 256, 0, stream>>>(out_w, Wt);
    k_inproj<<<(B_ * T_) / 16, 128, 0, stream>>>(x, in_w, in_b, xp_t);
    k_gru   <<<D_, 32, 0, stream>>>(xp_t, gru_wih, gru_whh, gru_bih, gru_bhh, gru_buf);
    k_static<<<1, 256, 0, stream>>>(stat, static_w, static_b, out_w, out_b, s_contrib);
    k_out   <<<(B_ * T_) / 32, 256, 0, stream>>>(gru_buf, Wt, s_contrib, out);
}